// PixelwiseSpectralAttentionPooling_3753801417445
// MI455X (gfx1250) — compile-verified
//
#include <hip/hip_runtime.h>
#include <math.h>

// Problem constants (from reference): N=4096, T=64, D=1024, fp32.
#define N_ 4096
#define T_ 64
#define D_ 1024
#define RS 1028   // padded LDS row stride in floats (1028 % 64 == 4 -> conflict-free WMMA A fetch)

typedef float v2f __attribute__((ext_vector_type(2)));
typedef float v8f __attribute__((ext_vector_type(8)));

// Async global->LDS copy (CDNA5 VFLAT async op, tracked by ASYNCcnt).
// vdst = 32-bit LDS byte offset (low 32 bits of a generic shared pointer),
// vaddr = 64-bit global address, GV addressing mode.
__device__ __forceinline__ void async_copy_b128(const float* gsrc, float* ldst) {
    unsigned int  loff = (unsigned int)(unsigned long long)ldst;
    unsigned long long ga = (unsigned long long)gsrc;
    asm volatile("global_load_async_to_lds_b128 %0, %1, off"
                 :: "v"(loff), "v"(ga) : "memory");
}

__device__ __forceinline__ void wait_async0() {
    asm volatile("s_wait_asynccnt 0" ::: "memory");
}

__global__ __launch_bounds__(256)
void attnpool_kernel(const float* __restrict__ h, const float* __restrict__ w,
                     float* __restrict__ pooled, float* __restrict__ alpha_out) {
    extern __shared__ float smem[];
    float* sh     = smem;              // 64 * RS floats (padded h tile)
    float* sw     = smem + T_ * RS;    // 1024 floats (w_score)
    float* sPart  = sw + D_;           // 128 floats (score partials, 2 D-halves x 64 t)
    float* sAlpha = sPart + 128;       // 64 floats

    const int tid   = threadIdx.x;
    const int n     = blockIdx.x;
    const int lane  = tid & 31;
    const int wave  = tid >> 5;
    const int laneM = lane & 15;   // WMMA row within half-wave
    const int kh    = lane >> 4;   // K-half select (ISA 16x4 f32 A layout)

    const float* hn = h + (size_t)n * (T_ * D_);

    // Stage w_score into LDS (256 threads x float4 = 1024 floats; L2-resident across blocks).
    ((float4*)sw)[tid] = ((const float4*)w)[tid];

    // Async-stage the whole 64x1024 fp32 h[n] tile into (padded) LDS: one B128 per thread per row.
    for (int r = 0; r < T_; ++r) {
        async_copy_b128(hn + r * D_ + tid * 4, sh + r * RS + tid * 4);
    }
    wait_async0();
    __syncthreads();

    // ---------------- Phase 1: scores[t] = dot(h[t,:], w) via V_WMMA_F32_16X16X4_F32 ----------------
    // Wave w handles t-tile (w&3) over D-half (w>>2). B columns are broadcast of w[d0+k],
    // so every column of C equals the score partial for row t.
    {
        const int ttile  = wave & 3;
        const int dh     = wave >> 2;
        const int rowoff = (ttile * 16 + laneM) * RS;
        const int dstart = dh * 512;
        v8f c = {};
        for (int d0 = dstart; d0 < dstart + 512; d0 += 4) {
            const int dk = d0 + 2 * kh;
            v2f a, b;
            a.x = sh[rowoff + dk];
            a.y = sh[rowoff + dk + 1];
            b.x = sw[dk];
            b.y = sw[dk + 1];
            c = __builtin_amdgcn_wmma_f32_16x16x4_f32(false, a, false, b, (short)0, c, false, false);
        }
        if (laneM == 0) {   // lane 0 holds rows 0..7 (col 0), lane 16 holds rows 8..15
            #pragma unroll
            for (int r = 0; r < 8; ++r)
                sPart[dh * 64 + ttile * 16 + kh * 8 + r] = c[r];
        }
    }
    __syncthreads();

    // ---------------- Phase 2: softmax over T=64 (redundant per-thread; trivial cost) ----------------
    float mx = -3.402823466e+38f;
    for (int t = 0; t < T_; ++t) {
        float s = sPart[t] + sPart[64 + t];
        mx = fmaxf(mx, s);
    }
    float sum = 0.0f;
    for (int t = 0; t < T_; ++t) {
        float s = sPart[t] + sPart[64 + t];
        sum += expf(s - mx);
    }
    const float inv = 1.0f / sum;
    if (tid < T_) {
        float s = sPart[tid] + sPart[64 + tid];
        float a = expf(s - mx) * inv;
        sAlpha[tid] = a;
        alpha_out[(size_t)n * T_ + tid] = a;
    }
    __syncthreads();

    // ---------------- Phase 3: pooled[d] = sum_t alpha[t]*h[t,d] via WMMA ----------------
    // A[m,k] = h[4j+k, d0+m] (transposed LDS fetch), B[k,n] = alpha[4j+k] broadcast.
    // C[m, any n] = pooled[d0+m]; accumulate over j = 0..15 (covers T=64).
    for (int i = 0; i < 8; ++i) {
        const int d0 = (wave * 8 + i) * 16;
        v8f c = {};
        #pragma unroll
        for (int j = 0; j < 16; ++j) {
            const int t0 = 4 * j + 2 * kh;
            v2f a, b;
            a.x = sh[t0 * RS + d0 + laneM];
            a.y = sh[(t0 + 1) * RS + d0 + laneM];
            b.x = sAlpha[t0];
            b.y = sAlpha[t0 + 1];
            c = __builtin_amdgcn_wmma_f32_16x16x4_f32(false, a, false, b, (short)0, c, false, false);
        }
        if (laneM == 0) {
            #pragma unroll
            for (int r = 0; r < 8; ++r)
                pooled[(size_t)n * D_ + d0 + kh * 8 + r] = c[r];
        }
    }
}

extern "C" void kernel_launch(void* const* d_in, const int* in_sizes, int n_in,
                              void* d_out, int out_size, void* d_ws, size_t ws_size,
                              hipStream_t stream) {
    const float* h = (const float*)d_in[0];       // [N, T, D] fp32
    const float* w = (const float*)d_in[1];       // [D] fp32
    float* out    = (float*)d_out;
    float* pooled = out;                          // [N, D]
    float* alpha  = out + (size_t)N_ * D_;        // [N, T]

    const size_t lds_bytes = (size_t)(T_ * RS + D_ + 128 + 64) * sizeof(float); // ~268 KB < 320 KB WGP LDS
    (void)hipFuncSetAttribute((const void*)attnpool_kernel,
                              hipFuncAttributeMaxDynamicSharedMemorySize, (int)lds_bytes);

    attnpool_kernel<<<N_, 256, lds_bytes, stream>>>(h, w, pooled, alpha);
}